// Detector_37735582663083
// MI455X (gfx1250) — compile-verified
//
#include <hip/hip_runtime.h>
#include <hip/hip_bf16.h>
#include <math.h>

#define H 128
#define W 128
#define NPIX (H * W)          // 16384
#define K 2048
#define SCORE_THRESH 0.8f
#define BOX_THRESH 0.1f
#define NMS_THRESH 0.2f

typedef __attribute__((ext_vector_type(2))) float v2f;
typedef __attribute__((ext_vector_type(8))) float v8f;

// ---------------------------------------------------------------- init
__global__ void east_init_kernel(float* __restrict__ val, int* __restrict__ idx) {
    int k = blockIdx.x * blockDim.x + threadIdx.x;
    if (k < K) { val[k] = -__builtin_inff(); idx[k] = 0; }
}

// ------------------------------------------------------- top-k by rank
// rank(t) = #{ valid j : s_j > s_t  ||  (s_j == s_t && j < t) }
// Reproduces jax.lax.top_k ordering (descending, ties -> smaller index).
__global__ void east_topk_kernel(const float* __restrict__ s,
                                 float* __restrict__ out_val,
                                 int* __restrict__ out_idx) {
    __shared__ float tile[4096];
    int t = blockIdx.x * blockDim.x + threadIdx.x;   // 0..16383
    float st = s[t];
    bool valid = st > SCORE_THRESH;
    int rank = 0;
    for (int c = 0; c < NPIX; c += 4096) {
        for (int i = threadIdx.x; i < 4096; i += blockDim.x) tile[i] = s[c + i];
        __syncthreads();
        if (valid) {
            for (int j = 0; j < 4096; ++j) {
                float sj = tile[j];
                int gj = c + j;
                if (sj > SCORE_THRESH && (sj > st || (sj == st && gj < t))) rank++;
            }
        }
        __syncthreads();
    }
    if (valid && rank < K) { out_val[rank] = st; out_idx[rank] = t; }
}

// --------------------------------------------- geometry / edge coeffs
__global__ void east_geom_kernel(const float* __restrict__ geo,
                                 const float* __restrict__ val,
                                 const int* __restrict__ idx,
                                 float* __restrict__ quads,
                                 float* __restrict__ xmn_o, float* __restrict__ xmx_o,
                                 float* __restrict__ ymn_o, float* __restrict__ ymx_o,
                                 float* __restrict__ area_o,
                                 float* __restrict__ coeff,
                                 int* __restrict__ keep) {
    int k = blockIdx.x * blockDim.x + threadIdx.x;
    if (k >= K) return;
    float v = val[k];
    bool valid = v > SCORE_THRESH;
    int id = idx[k];
    int row = id >> 7, col = id & (W - 1);
    float ox = (float)col * 4.0f, oy = (float)row * 4.0f;
    const float* g = geo + (size_t)id * 5;
    float d0 = g[0], d1 = g[1], d2 = g[2], d3 = g[3], ang = g[4];
    float w = d1 + d3, h = d0 + d2;

    float px[5], py[5], rx[5], ry[5];
    if (ang >= 0.0f) {
        float ca = cosf(ang), sa = sinf(ang);
        px[0] = 0;  py[0] = -h;  px[1] = w;   py[1] = -h;
        px[2] = w;  py[2] = 0;   px[3] = 0;   py[3] = 0;
        px[4] = d3; py[4] = -d2;
        for (int j = 0; j < 5; ++j) { rx[j] =  ca * px[j] + sa * py[j];
                                      ry[j] = -sa * px[j] + ca * py[j]; }
    } else {
        float cb = cosf(-ang), sb = sinf(-ang);
        px[0] = -w;  py[0] = -h;  px[1] = 0;  py[1] = -h;
        px[2] = 0;   py[2] = 0;   px[3] = -w; py[3] = 0;
        px[4] = -d1; py[4] = -d2;
        for (int j = 0; j < 5; ++j) { rx[j] = cb * px[j] - sb * py[j];
                                      ry[j] = sb * px[j] + cb * py[j]; }
    }
    float shx = ox - rx[4], shy = oy - ry[4];
    float qx[4], qy[4];
    float xmn = 1e30f, xmx = -1e30f, ymn = 1e30f, ymx = -1e30f;
    for (int j = 0; j < 4; ++j) {
        qx[j] = rx[j] + shx; qy[j] = ry[j] + shy;
        quads[k * 8 + j * 2 + 0] = qx[j];
        quads[k * 8 + j * 2 + 1] = qy[j];
        xmn = fminf(xmn, qx[j]); xmx = fmaxf(xmx, qx[j]);
        ymn = fminf(ymn, qy[j]); ymx = fmaxf(ymx, qy[j]);
    }
    // floored /4 coordinates -> integer-valued floats; edge affine coeffs:
    // cr(px,py) = cA*py + cB*px + cC  ==  (x2-x1)(py-y1) - (y2-y1)(px-x1)
    float fx[4], fy[4];
    for (int j = 0; j < 4; ++j) { fx[j] = floorf(qx[j] * 0.25f); fy[j] = floorf(qy[j] * 0.25f); }
    for (int e = 0; e < 4; ++e) {
        float x1 = fx[e], y1 = fy[e], x2 = fx[(e + 1) & 3], y2 = fy[(e + 1) & 3];
        float cA = x2 - x1;
        float cB = -(y2 - y1);
        float cC = (y2 - y1) * x1 - (x2 - x1) * y1;
        if (!valid) { cA = 0.f; cB = 0.f; cC = 0.f; }
        float* cp = coeff + (size_t)k * 16 + e * 4;
        cp[0] = cA; cp[1] = cB; cp[2] = cC; cp[3] = 0.0f;
    }
    xmn_o[k] = valid ? xmn : 0.f;  xmx_o[k] = valid ? xmx : 0.f;
    ymn_o[k] = valid ? ymn : 0.f;  ymx_o[k] = valid ? ymx : 0.f;
    area_o[k] = valid ? (xmx - xmn) * (ymx - ymn) : 0.f;
    keep[k] = valid ? 1 : 0;
}

// ------------------------------------------------- sequential-scan NMS
__global__ void east_nms_kernel(const float* __restrict__ xmn, const float* __restrict__ xmx,
                                const float* __restrict__ ymn, const float* __restrict__ ymx,
                                const float* __restrict__ area, int* __restrict__ keep) {
    __shared__ float sxn[K], sxx[K], syn[K], syx[K], sar[K];
    __shared__ int skeep[K];
    for (int i = threadIdx.x; i < K; i += blockDim.x) {
        sxn[i] = xmn[i]; sxx[i] = xmx[i]; syn[i] = ymn[i]; syx[i] = ymx[i];
        sar[i] = area[i]; skeep[i] = keep[i];
    }
    __syncthreads();
    for (int i = 0; i < K - 1; ++i) {
        if (skeep[i]) {
            float axn = sxn[i], axx = sxx[i], ayn = syn[i], ayx = syx[i], aar = sar[i];
            for (int j = i + 1 + threadIdx.x; j < K; j += blockDim.x) {
                if (skeep[j]) {
                    float ix = fmaxf(fminf(axx, sxx[j]) - fmaxf(axn, sxn[j]), 0.0f);
                    float iy = fmaxf(fminf(ayx, syx[j]) - fmaxf(ayn, syn[j]), 0.0f);
                    float inter = ix * iy;
                    float iou = inter / (aar + sar[j] - inter + 1e-8f);
                    if (iou > NMS_THRESH) skeep[j] = 0;
                }
            }
        }
        __syncthreads();
    }
    for (int i = threadIdx.x; i < K; i += blockDim.x) keep[i] = skeep[i];
}

// --------------------------------------- WMMA quad rasterization
// One wave handles 4 boxes (16 edges). Per pixel-tile of 16 pixels:
//   D[16x16] = A[16 edges x 4] * B[4 x 16 pixels],  B rows = [py; px; 1; 0]
// A layout (32-bit A 16x4): lanes 0-15 hold K=0 (v0) and K=1 (v1) of row M=lane;
// lanes 16-31 hold K=2 (v0) and K=3 (v1).
// B layout (row striped across lanes, C/D-style half split): v0 = rows K=0|K=2,
// v1 = rows K=1|K=3 across lane halves.
// D layout: lane n<16: v[j]=D[j][n]; lane 16+n: v[j]=D[j+8][n].
// Inside-test:  all(cr>=0) <=> min4(cr)>=0 ;  all(cr<=0) <=> max4(cr)<=0
// (min/max trees lower to v_min_num/v_max_num — 6 ops + 2 cmps per box,
//  vs 8 cmps + 6 logical ops for the naive form).
__global__ void east_raster_kernel(const float* __restrict__ s,
                                   const float* __restrict__ coeff,
                                   float* __restrict__ cnt,
                                   float* __restrict__ msum) {
    int wid = (blockIdx.x * blockDim.x + threadIdx.x) >> 5;  // wave id: 0..511
    int lane = threadIdx.x & 31;
    bool lo = lane < 16;
    int n = lane & 15;

    const float* ap = coeff + (size_t)wid * 64 + n * 4 + (lo ? 0 : 2);
    v2f a; a.x = ap[0]; a.y = ap[1];

    float c0 = 0.f, m0 = 0.f, c1 = 0.f, m1 = 0.f;
    for (int tt = 0; tt < (NPIX / 16); ++tt) {
        int py = tt >> 3;                 // 8 tiles of 16 pixels per row
        int pxb = (tt & 7) << 4;
        v2f b;
        b.x = lo ? (float)py : 1.0f;          // rows K=0 (py) | K=2 (ones)
        b.y = lo ? (float)(pxb + n) : 0.0f;   // rows K=1 (px) | K=3 (zeros)
        v8f c = {};
        v8f d = __builtin_amdgcn_wmma_f32_16x16x4_f32(
            false, a, false, b, (short)0, c, false, false);

        float sp = s[(py << 7) + pxb + n];

        float mnA = fminf(fminf(d[0], d[1]), fminf(d[2], d[3]));
        float mxA = fmaxf(fmaxf(d[0], d[1]), fmaxf(d[2], d[3]));
        float mnB = fminf(fminf(d[4], d[5]), fminf(d[6], d[7]));
        float mxB = fmaxf(fmaxf(d[4], d[5]), fmaxf(d[6], d[7]));
        float mA = (mnA >= 0.0f || mxA <= 0.0f) ? 1.0f : 0.0f;
        float mB = (mnB >= 0.0f || mxB <= 0.0f) ? 1.0f : 0.0f;
        c0 += mA; m0 += mA * sp;
        c1 += mB; m1 += mB * sp;
    }
    // reduce within each 16-lane half (wave32): lanes 0-15 own boxes 0,1;
    // lanes 16-31 own boxes 2,3.
    for (int off = 8; off >= 1; off >>= 1) {
        c0 += __shfl_xor(c0, off, 32);
        m0 += __shfl_xor(m0, off, 32);
        c1 += __shfl_xor(c1, off, 32);
        m1 += __shfl_xor(m1, off, 32);
    }
    int gb = wid * 4;
    if (lane == 0)  { cnt[gb + 0] = c0; msum[gb + 0] = m0; cnt[gb + 1] = c1; msum[gb + 1] = m1; }
    if (lane == 16) { cnt[gb + 2] = c0; msum[gb + 2] = m0; cnt[gb + 3] = c1; msum[gb + 3] = m1; }
}

// ---------------------------------------------------------- finalize
__global__ void east_final_kernel(const float* __restrict__ quads,
                                  const float* __restrict__ cnt,
                                  const float* __restrict__ msum,
                                  const int* __restrict__ keep,
                                  float* __restrict__ out) {
    int k = blockIdx.x * blockDim.x + threadIdx.x;
    if (k >= K) return;
    float mean = msum[k] / fmaxf(cnt[k], 1.0f);
    bool f = (keep[k] != 0) && (mean > BOX_THRESH);
    float* o = out + (size_t)k * 9;
    for (int j = 0; j < 8; ++j) o[j] = f ? quads[k * 8 + j] : 0.0f;
    o[8] = f ? mean : 0.0f;
}

// ------------------------------------------------------------ launch
extern "C" void kernel_launch(void* const* d_in, const int* in_sizes, int n_in,
                              void* d_out, int out_size, void* d_ws, size_t ws_size,
                              hipStream_t stream) {
    const float* score = (const float*)d_in[0];  // [1,128,128,1]
    const float* geo   = (const float*)d_in[1];  // [1,128,128,5]
    float* out = (float*)d_out;                  // [K,9]

    // workspace layout (floats); total = 34*K floats = 278528 bytes
    float* wsf   = (float*)d_ws;
    float* wval  = wsf;                 // K
    int*   widx  = (int*)(wsf + K);     // K
    float* quads = wsf + 2 * K;         // 8K
    float* xmn   = quads + 8 * K;       // K
    float* xmx   = xmn + K;             // K
    float* ymn   = xmx + K;             // K
    float* ymx   = ymn + K;             // K
    float* ar    = ymx + K;             // K
    float* coeff = ar + K;              // 16K (byte offset 122880, 8B-aligned)
    int*   keep  = (int*)(coeff + 16 * K); // K
    float* cnt   = (float*)(keep + K);  // K
    float* msm   = cnt + K;             // K
    (void)ws_size; (void)in_sizes; (void)n_in; (void)out_size;

    east_init_kernel<<<(K + 255) / 256, 256, 0, stream>>>(wval, widx);
    east_topk_kernel<<<NPIX / 256, 256, 0, stream>>>(score, wval, widx);
    east_geom_kernel<<<K / 256, 256, 0, stream>>>(geo, wval, widx, quads,
                                                  xmn, xmx, ymn, ymx, ar, coeff, keep);
    east_nms_kernel<<<1, 1024, 0, stream>>>(xmn, xmx, ymn, ymx, ar, keep);
    east_raster_kernel<<<64, 256, 0, stream>>>(score, coeff, cnt, msm);  // 512 waves = 2048 boxes
    east_final_kernel<<<K / 256, 256, 0, stream>>>(quads, cnt, msm, keep, out);
}